// Profile_38611755991459
// MI455X (gfx1250) — compile-verified
//
#include <hip/hip_runtime.h>
#include <math.h>

typedef __attribute__((ext_vector_type(2))) float v2f;
typedef __attribute__((ext_vector_type(8))) float v8f;

#define Bn 4096
#define Dn 2048
#define Kn 5
#define Pn 1323
#define NCOL 47           // 5 + 12 + 30
#define NPAD 48           // padded to 3 WMMA tiles
#define LOG_2PI 1.8378770664093453f
#define LOG2E   1.4426950408889634f
#define PROF_ELEMS (Bn * Pn)     // profile output elements, L follows

// ---------------------------------------------------------------------------
// Kernel 1: S = rep @ [W_mix;W_mean;W_scale]^T   via V_WMMA_F32_16X16X4_F32
// One wave computes a 16x48 strip (3 accumulators share each A fragment, so
// `rep` is streamed from HBM exactly once).
// ---------------------------------------------------------------------------
__global__ __launch_bounds__(128) void gemm_wmma_kernel(
    const float* __restrict__ rep,
    const float* __restrict__ Wmix,
    const float* __restrict__ Wmean,
    const float* __restrict__ Wscale,
    float* __restrict__ Sws) {
  const int lane  = threadIdx.x & 31;
  const int mTile = blockIdx.x * 4 + (threadIdx.x >> 5);   // 256 tiles of 16 rows
  const int mLane = lane & 15;
  const int koff  = (lane >> 4) * 2;   // lanes 0-15: K 0,1 ; lanes 16-31: K 2,3

  // A: per-lane row pointer into rep
  const float* arow = rep + (size_t)(mTile * 16 + mLane) * Dn;

  // B: per-lane weight-row pointers for the 3 column tiles
  const float* wrow[3];
#pragma unroll
  for (int t = 0; t < 3; ++t) {
    int n = t * 16 + mLane;
    const float* p;
    if (n < 5)       p = Wmix   + (size_t)n * Dn;
    else if (n < 17) p = Wmean  + (size_t)(n - 5) * Dn;
    else if (n < 47) p = Wscale + (size_t)(n - 17) * Dn;
    else             p = Wmix;                 // pad column 47: never consumed
    wrow[t] = p;
  }

  v8f acc[3] = {v8f{}, v8f{}, v8f{}};
#pragma unroll 8
  for (int kk = 0; kk < Dn; kk += 4) {
    v2f a = *(const v2f*)(arow + kk + koff);
#pragma unroll
    for (int t = 0; t < 3; ++t) {
      v2f bf = *(const v2f*)(wrow[t] + kk + koff);
      acc[t] = __builtin_amdgcn_wmma_f32_16x16x4_f32(
          false, a, false, bf, (short)0, acc[t], false, false);
    }
  }

  // C/D layout: VGPR r -> (M = r + (lane>=16 ? 8 : 0), N = lane&15)
  const int rBase = (lane >> 4) * 8;
#pragma unroll
  for (int t = 0; t < 3; ++t) {
#pragma unroll
    for (int r = 0; r < 8; ++r) {
      int row = mTile * 16 + rBase + r;
      Sws[(size_t)row * NPAD + t * 16 + mLane] = acc[t][r];
    }
  }
}

// ---------------------------------------------------------------------------
// Kernel 2: per-b parameter prep + emit L output.
// Packs per (b,k): [mx,my,mz, 1/L00, L10, 1/L11, L20, L21, 1/L22, c2]
// where c2 = (log_w - log_det - 1.5*log(2pi)) * log2(e).
// ---------------------------------------------------------------------------
__device__ __forceinline__ float softplus_f(float x) {
  return fmaxf(x, 0.0f) + log1pf(expf(-fabsf(x)));
}

__global__ __launch_bounds__(256) void params_kernel(
    const float* __restrict__ Sws,
    const float* __restrict__ bmix,
    const float* __restrict__ bmean,
    const float* __restrict__ bscale,
    float* __restrict__ Pws,          // [B][52] padded rows
    float* __restrict__ out) {        // L goes at out + PROF_ELEMS
  const int b = blockIdx.x * blockDim.x + threadIdx.x;
  if (b >= Bn) return;
  const float* Sr = Sws + (size_t)b * NPAD;

  // log_softmax over mixture logits
  float mix[Kn];
  float mmax = -INFINITY;
#pragma unroll
  for (int k = 0; k < Kn; ++k) { mix[k] = Sr[k] + bmix[k]; mmax = fmaxf(mmax, mix[k]); }
  float se = 0.0f;
#pragma unroll
  for (int k = 0; k < Kn; ++k) se += expf(mix[k] - mmax);
  const float lse = mmax + logf(se);

  float* prow = Pws + (size_t)b * 52;
  float* Lout = out + (size_t)PROF_ELEMS + (size_t)b * (Kn * 9);

#pragma unroll
  for (int k = 0; k < Kn; ++k) {
    float mx = 0.0f, my = 0.0f, mz = 0.0f;
    if (k > 0) {
      int o = (k - 1) * 3;
      mx = Sr[5 + o + 0] + bmean[o + 0];
      my = Sr[5 + o + 1] + bmean[o + 1];
      mz = Sr[5 + o + 2] + bmean[o + 2];
    }
    const int so = 17 + k * 6;
    const int bo = k * 6;
    float L00 = softplus_f(Sr[so + 0] + bscale[bo + 0]);
    float L10 = Sr[so + 1] + bscale[bo + 1];
    float L11 = softplus_f(Sr[so + 2] + bscale[bo + 2]);
    float L20 = Sr[so + 3] + bscale[bo + 3];
    float L21 = Sr[so + 4] + bscale[bo + 4];
    float L22 = softplus_f(Sr[so + 5] + bscale[bo + 5]);

    float logdet = logf(L00) + logf(L11) + logf(L22);
    float lw = mix[k] - lse;
    float c2 = (lw - logdet - 1.5f * LOG_2PI) * LOG2E;

    float* pk = prow + k * 10;
    pk[0] = mx;  pk[1] = my;  pk[2] = mz;
    pk[3] = 1.0f / L00;  pk[4] = L10;  pk[5] = 1.0f / L11;
    pk[6] = L20; pk[7] = L21; pk[8] = 1.0f / L22; pk[9] = c2;

    float* lk = Lout + k * 9;
    lk[0] = L00; lk[1] = 0.0f; lk[2] = 0.0f;
    lk[3] = L10; lk[4] = L11;  lk[5] = 0.0f;
    lk[6] = L20; lk[7] = L21;  lk[8] = L22;
  }
}

// ---------------------------------------------------------------------------
// Kernel 3: streaming mixture evaluation. Lane owns one b (params in regs),
// walks 21 consecutive p values. out flat index = p*B + b (the reference's
// reshape is a pure reinterpret of the [P,B] array).
// ---------------------------------------------------------------------------
#define P_PER 21   // 1323 / 63 chunks

__global__ __launch_bounds__(256) void mixture_kernel(
    const float* __restrict__ dxyz,   // [P][B][3]
    const float* __restrict__ Pws,    // [B][52]
    float* __restrict__ out) {
  const int b = blockIdx.x * blockDim.x + threadIdx.x;
  const int pStart = blockIdx.y * P_PER;

  float pr[Kn][10];
  const float* prow = Pws + (size_t)b * 52;
#pragma unroll
  for (int k = 0; k < Kn; ++k)
#pragma unroll
    for (int j = 0; j < 10; ++j) pr[k][j] = prow[k * 10 + j];

  for (int i = 0; i < P_PER; ++i) {
    const int p = pStart + i;
    const size_t base = (size_t)p * Bn + b;
    const float* dp = dxyz + base * 3;
    float x = dp[0], y = dp[1], z = dp[2];
    // prefetch next p-row for this lane (global_prefetch_b8)
    __builtin_prefetch(dp + (size_t)3 * Bn, 0, 1);

    float lp2[Kn];
    float m2 = -INFINITY;
#pragma unroll
    for (int k = 0; k < Kn; ++k) {
      float dx = x - pr[k][0];
      float dy = y - pr[k][1];
      float dz = z - pr[k][2];
      float z0 = dx * pr[k][3];
      float z1 = (dy - pr[k][4] * z0) * pr[k][5];
      float z2 = (dz - pr[k][6] * z0 - pr[k][7] * z1) * pr[k][8];
      float maha = fmaf(z0, z0, fmaf(z1, z1, z2 * z2));
      lp2[k] = fmaf(maha, -0.5f * LOG2E, pr[k][9]);
      m2 = fmaxf(m2, lp2[k]);
    }
    float s = 0.0f;
#pragma unroll
    for (int k = 0; k < Kn; ++k) s += exp2f(lp2[k] - m2);
    out[base] = exp2f(m2) * s;
  }
}

// ---------------------------------------------------------------------------
extern "C" void kernel_launch(void* const* d_in, const int* in_sizes, int n_in,
                              void* d_out, int out_size, void* d_ws, size_t ws_size,
                              hipStream_t stream) {
  (void)in_sizes; (void)n_in; (void)out_size; (void)ws_size;
  const float* rep    = (const float*)d_in[0];
  const float* dxyz   = (const float*)d_in[1];
  const float* Wmix   = (const float*)d_in[2];
  const float* bmix   = (const float*)d_in[3];
  const float* Wmean  = (const float*)d_in[4];
  const float* bmean  = (const float*)d_in[5];
  const float* Wscale = (const float*)d_in[6];
  const float* bscale = (const float*)d_in[7];
  float* out = (float*)d_out;

  float* Sws = (float*)d_ws;                                   // [B][48]
  float* Pws = (float*)((char*)d_ws + (size_t)Bn * NPAD * 4);  // [B][52]

  // Phase 1: WMMA GEMM (256 waves, 4 waves/block)
  gemm_wmma_kernel<<<dim3(Bn / 64), dim3(128), 0, stream>>>(rep, Wmix, Wmean, Wscale, Sws);

  // Phase 2: parameter prep + L output
  params_kernel<<<dim3(Bn / 256), dim3(256), 0, stream>>>(Sws, bmix, bmean, bscale, Pws, out);

  // Phase 3: streaming mixture evaluation (16 x 63 blocks)
  mixture_kernel<<<dim3(Bn / 256, Pn / P_PER), dim3(256), 0, stream>>>(dxyz, Pws, out);
}